// SinkhornNormalizer_88244398064099
// MI455X (gfx1250) — compile-verified
//
#include <hip/hip_runtime.h>
#include <cstdint>
#include <cstddef>

// ---------------------------------------------------------------------------
// Sinkhorn-Knopp (log domain) for logits[64][1024][1024] f32, 5 iterations.
// State kept as potentials u[b][i], v[b][j]:
//   u_i = LSE_j(logits_ij - v_j)        (row pass)
//   v_j = LSE_i(logits_ij - u_i)        (col pass, max-free: row-normalized)
//   out = exp(logits - u_i - v_j)
// Col pass: 16x256 f32 tiles DMA'd into LDS by the Tensor Data Mover
// (tensor_load_to_lds, TENSORcnt), double-buffered; inner loop is one
// v_fma + v_exp per element. Row/output passes stage the v vector with
// global_load_async_to_lds_b128 (ASYNCcnt).
// ---------------------------------------------------------------------------

#define SK_B 64
#define SK_N 1024
#define SK_ITERS 5

#define LOG2E 1.4426950408889634f
#define LN2 0.6931471805599453f

typedef float v4f __attribute__((ext_vector_type(4)));
typedef unsigned u32x4 __attribute__((ext_vector_type(4)));
typedef unsigned u32x8 __attribute__((ext_vector_type(8)));

__device__ __forceinline__ float exp2_raw(float x) {
  return __builtin_amdgcn_exp2f(x);                // v_exp_f32
}
__device__ __forceinline__ float fast_log(float x) {
  return __builtin_amdgcn_logf(x) * LN2;           // v_log_f32
}

// Flat pointer to __shared__ -> 32-bit LDS offset (low 32 bits of aperture).
__device__ __forceinline__ unsigned lds_off(const void* p) {
  return (unsigned)(uintptr_t)p;
}

// CDNA5 async global->LDS copy (ASYNCcnt-tracked), 128 bits per lane.
__device__ __forceinline__ void async_ld_b128(unsigned lds_byte_off,
                                              const float* gaddr) {
  asm volatile("global_load_async_to_lds_b128 %0, %1, off"
               :: "v"(lds_byte_off), "v"(gaddr)
               : "memory");
}
#if defined(__has_builtin) && __has_builtin(__builtin_amdgcn_s_wait_asynccnt)
#define WAIT_ASYNC(n) __builtin_amdgcn_s_wait_asynccnt(n)
#else
#define WAIT_ASYNC(n) asm volatile("s_wait_asynccnt " #n ::: "memory")
#endif
#if defined(__has_builtin) && __has_builtin(__builtin_amdgcn_s_wait_tensorcnt)
#define WAIT_TENSOR(n) __builtin_amdgcn_s_wait_tensorcnt(n)
#else
#define WAIT_TENSOR(n) asm volatile("s_wait_tensorcnt " #n ::: "memory")
#endif

// Tensor Data Mover: DMA one 2D tile (16 rows x 256 f32, row stride SK_N
// elems) from global into LDS at lds_byte. One instruction per wave;
// tracked with TENSORcnt. D# layout per CDNA5 ISA ch.8.
__device__ __forceinline__ void tdm_load_tile_f32(const float* gptr,
                                                  unsigned lds_byte) {
  const uint64_t ga = (uint64_t)(uintptr_t)gptr;
  u32x4 g0;
  g0.x = 1u;                                    // count=1, user descriptor
  g0.y = lds_byte;                              // lds_addr[31:0]
  g0.z = (unsigned)ga;                          // global_addr[31:0]
  g0.w = ((unsigned)(ga >> 32) & 0x01FFFFFFu)   // global_addr[56:32]
         | 0x80000000u;                         // type=2 ("image")
  u32x8 g1;
  g1.s0 = 2u << 16;                // workgroup_mask=0, data_size=2 (4B)
  g1.s1 = (SK_N & 0xFFFFu) << 16;  // tensor_dim0 = 1024 (low 16)
  g1.s2 = (SK_N & 0xFFFFu) << 16;  // tensor_dim0 hi=0 | tensor_dim1 low16
  g1.s3 = 256u << 16;              // tensor_dim1 hi=0 | tile_dim0 = 256
  g1.s4 = 16u;                     // tile_dim1 = 16, tile_dim2 = 0
  g1.s5 = (unsigned)SK_N;          // tensor_dim0_stride = 1024 (low 32)
  g1.s6 = 0u;                      // dim0_stride hi | dim1_stride low16
  g1.s7 = (SK_N * SK_N) >> 16;     // tensor_dim1_stride hi (unused: 2D tile)
  asm volatile("tensor_load_to_lds %0, %1" :: "s"(g0), "s"(g1) : "memory");
}

// --------------------------------------------------------------------- init
__global__ void sk_init_v(float* __restrict__ v) {
  int idx = blockIdx.x * blockDim.x + threadIdx.x;
  if (idx < SK_B * SK_N) v[idx] = 0.0f;
}

// ----------------------------------------------------------------- row pass
// One wave per row (32 lanes x 32 elements, all in VGPRs). 8 rows / block.
__global__ void sk_row_pass(const float* __restrict__ logits,
                            const float* __restrict__ v,
                            float* __restrict__ u) {
  const int b = blockIdx.x;
  const int t = threadIdx.x;
  const int lane = t & 31;
  const int wv = t >> 5;
  const int row = blockIdx.y * 8 + wv;

  __shared__ __align__(16) float sv[SK_N];
  // Stage v[b][:] into LDS with the async copy engine (no VGPR round-trip).
  async_ld_b128(lds_off(&sv[t * 4]), v + (size_t)b * SK_N + t * 4);
  WAIT_ASYNC(0);
  __syncthreads();

  const v4f* rp = reinterpret_cast<const v4f*>(
      logits + ((size_t)b * SK_N + row) * SK_N);
  const v4f* sv4 = reinterpret_cast<const v4f*>(sv);

  v4f d[8];
#pragma unroll
  for (int c = 0; c < 8; ++c) d[c] = rp[c * 32 + lane] - sv4[c * 32 + lane];

  float m = -__builtin_inff();
#pragma unroll
  for (int c = 0; c < 8; ++c) {
    m = fmaxf(m, fmaxf(fmaxf(d[c].x, d[c].y), fmaxf(d[c].z, d[c].w)));
  }
#pragma unroll
  for (int off = 16; off > 0; off >>= 1) m = fmaxf(m, __shfl_xor(m, off, 32));

  const float nmL = -m * LOG2E;  // exp(d-m) = exp2(d*log2e - m*log2e)
  float s = 0.0f;
#pragma unroll
  for (int c = 0; c < 8; ++c) {
    s += exp2_raw(fmaf(d[c].x, LOG2E, nmL)) +
         exp2_raw(fmaf(d[c].y, LOG2E, nmL)) +
         exp2_raw(fmaf(d[c].z, LOG2E, nmL)) +
         exp2_raw(fmaf(d[c].w, LOG2E, nmL));
  }
#pragma unroll
  for (int off = 16; off > 0; off >>= 1) s += __shfl_xor(s, off, 32);

  if (lane == 0) u[(size_t)b * SK_N + row] = m + fast_log(s);
}

// ----------------------------------------------------------------- col pass
// Block = 256 columns x 128 rows. Tiles of 16 rows x 256 cols are DMA'd into
// LDS by the Tensor Data Mover (wave 0 issues; TENSORcnt pipelined, double
// buffered). u values are block-uniform scalar loads folded into the exp2
// bias, so the per-element work is exactly v_fma_f32 + v_exp_f32.
__global__ void sk_col_pass(const float* __restrict__ logits,
                            const float* __restrict__ u,
                            float* __restrict__ partial) {
  constexpr int TR = 16;      // tile rows
  constexpr int JB = 256;     // cols per block
  constexpr int CHUNK = 128;  // rows per block
  constexpr int TILES = CHUNK / TR;

  const int b = blockIdx.x;
  const int jblk = blockIdx.y;
  const int ich = blockIdx.z;
  const int t = threadIdx.x;
  const bool issuer = (t < 32);  // wave 0 drives the TDM

  __shared__ __align__(16) float tiles[2 * TR * JB];  // 32 KB double buffer

  const int i0 = ich * CHUNK;
  const float* gbase = logits + ((size_t)b * SK_N + i0) * SK_N + jblk * JB;
  const float* ug = u + (size_t)b * SK_N + i0;  // block-uniform (scalar loads)

  if (issuer) {
    tdm_load_tile_f32(gbase, lds_off(&tiles[0]));
  }

  float s = 0.0f;
  for (int k = 0; k < TILES; ++k) {
    if (issuer) {
      if (k + 1 < TILES) {
        tdm_load_tile_f32(gbase + (size_t)(k + 1) * TR * SK_N,
                          lds_off(&tiles[((k + 1) & 1) * TR * JB]));
        WAIT_TENSOR(1);  // tile k landed (in-order); tile k+1 in flight
      } else {
        WAIT_TENSOR(0);
      }
    }
    __syncthreads();   // publish tile k to all waves
    const float* tb = tiles + (k & 1) * TR * JB;
    float nu[TR];      // -u[i]*log2e, uniform per block
#pragma unroll
    for (int r = 0; r < TR; ++r) nu[r] = ug[k * TR + r] * (-LOG2E);
#pragma unroll
    for (int r = 0; r < TR; ++r)
      s += exp2_raw(fmaf(tb[r * JB + t], LOG2E, nu[r]));
    __syncthreads();   // everyone done reading buf before TDM refills it
  }

  partial[((size_t)b * (SK_N / CHUNK) + ich) * SK_N + jblk * JB + t] = s;
}

// ------------------------------------------------------------ finalize v
__global__ void sk_finalize_v(const float* __restrict__ partial,
                              float* __restrict__ v) {
  const int idx = blockIdx.x * blockDim.x + threadIdx.x;  // B*N threads
  const int b = idx >> 10;
  const int j = idx & (SK_N - 1);
  float s = 0.0f;
#pragma unroll
  for (int c = 0; c < 8; ++c)
    s += partial[((size_t)b * 8 + c) * SK_N + j];
  v[idx] = fast_log(s);
}

// --------------------------------------------------------------- output
__global__ void sk_output(const float* __restrict__ logits,
                          const float* __restrict__ u,
                          const float* __restrict__ v,
                          float* __restrict__ out) {
  const int b = blockIdx.x;
  const int t = threadIdx.x;
  const int lane = t & 31;
  const int wv = t >> 5;
  const int row = blockIdx.y * 8 + wv;

  __shared__ __align__(16) float sv[SK_N];
  async_ld_b128(lds_off(&sv[t * 4]), v + (size_t)b * SK_N + t * 4);
  WAIT_ASYNC(0);
  __syncthreads();

  const float nuL = u[(size_t)b * SK_N + row] * (-LOG2E);  // wave-uniform
  const v4f* rp = reinterpret_cast<const v4f*>(
      logits + ((size_t)b * SK_N + row) * SK_N);
  v4f* op = reinterpret_cast<v4f*>(out + ((size_t)b * SK_N + row) * SK_N);
  const v4f* sv4 = reinterpret_cast<const v4f*>(sv);

#pragma unroll
  for (int c = 0; c < 8; ++c) {
    const v4f x = __builtin_nontemporal_load(&rp[c * 32 + lane]);  // last use
    const v4f vv = sv4[c * 32 + lane];
    v4f r;  // exp(x-u-v) = exp2(x*log2e + (-u*log2e - v*log2e)), 2 fma + exp
    r.x = exp2_raw(fmaf(x.x, LOG2E, fmaf(vv.x, -LOG2E, nuL)));
    r.y = exp2_raw(fmaf(x.y, LOG2E, fmaf(vv.y, -LOG2E, nuL)));
    r.z = exp2_raw(fmaf(x.z, LOG2E, fmaf(vv.z, -LOG2E, nuL)));
    r.w = exp2_raw(fmaf(x.w, LOG2E, fmaf(vv.w, -LOG2E, nuL)));
    __builtin_nontemporal_store(r, &op[c * 32 + lane]);  // streaming store
  }
}

// ---------------------------------------------------------------------------
extern "C" void kernel_launch(void* const* d_in, const int* in_sizes, int n_in,
                              void* d_out, int out_size, void* d_ws,
                              size_t ws_size, hipStream_t stream) {
  (void)in_sizes; (void)n_in; (void)out_size; (void)ws_size;
  const float* logits = (const float*)d_in[0];
  float* out = (float*)d_out;

  float* ws = (float*)d_ws;
  float* u = ws;                                   // B*N
  float* v = ws + (size_t)SK_B * SK_N;             // B*N
  float* partial = ws + (size_t)2 * SK_B * SK_N;   // B*8*N (col partials)

  sk_init_v<<<(SK_B * SK_N) / 256, 256, 0, stream>>>(v);

  for (int it = 0; it < SK_ITERS; ++it) {
    sk_row_pass<<<dim3(SK_B, SK_N / 8), 256, 0, stream>>>(logits, v, u);
    sk_col_pass<<<dim3(SK_B, SK_N / 256, SK_N / 128), 256, 0, stream>>>(
        logits, u, partial);
    sk_finalize_v<<<(SK_B * SK_N) / 256, 256, 0, stream>>>(partial, v);
  }

  sk_output<<<dim3(SK_B, SK_N / 8), 256, 0, stream>>>(logits, u, v, out);
}